// AttentionMechanism_52020643889803
// MI455X (gfx1250) — compile-verified
//
#include <hip/hip_runtime.h>

typedef float v2f __attribute__((ext_vector_type(2)));
typedef float v4f __attribute__((ext_vector_type(4)));
typedef float v8f __attribute__((ext_vector_type(8)));

#define NEG_SLOPE 0.2f
#define N_ROWS 8192
#define D_DIM 64

// ---------------------------------------------------------------------------
// Phase A: [s1|s2] = Wh (8192x64) @ [a1|a2] (64x2), via V_WMMA_F32_16X16X4_F32.
// One wave handles a 16-row tile; K=64 consumed in 16 WMMA steps of K=4.
//
// ISA layouts (CDNA5 ISA):
//   A (16x4 f32): lane L holds row M=L&15; VGPR0/1 = K {0,1} for lanes 0-15,
//                 K {2,3} for lanes 16-31.
//   B (4x16 f32): lane L holds col N=L&15; VGPR0/1 = rows K {0,1} (lanes 0-15),
//                 rows K {2,3} (lanes 16-31).
//   C/D (16x16):  VGPR v: lanes 0-15 -> (M=v, N=lane); lanes 16-31 -> (M=v+8).
//
// B construction is fully branchless: per-lane base pointer selected once
// (col 0 reads a1, col 1 reads a2, cols >=2 read a1 but are masked to zero by
// a per-lane 0/1 multiplier), so there is no EXEC manipulation in the loop and
// the a/Wh loads pipeline freely.
// ---------------------------------------------------------------------------
__global__ __launch_bounds__(256) void gat_s_kernel(const float* __restrict__ Wh,
                                                    const float* __restrict__ a,
                                                    float* __restrict__ s1,
                                                    float* __restrict__ s2) {
  const int lane  = threadIdx.x & 31;
  const int wave  = threadIdx.x >> 5;
  const int tile  = blockIdx.x * 8 + wave;   // 512 tiles of 16 rows
  const int row16 = tile * 16;
  const int m     = lane & 15;               // row (A) / col (B) within tile
  const int khalf = (lane >> 4) * 2;         // 0 for lanes 0-15, 2 for 16-31

  const float* __restrict__ a1 = a;          // a[:64]
  const float* __restrict__ a2 = a + D_DIM;  // a[64:]
  const float* rowPtr = Wh + (size_t)(row16 + m) * D_DIM;

  // Select B source once per lane; mask kills columns >= 2.
  const float* bp    = (m == 1) ? a2 : a1;
  const float  bmask = (m < 2) ? 1.0f : 0.0f;

  v8f c = {};
  #pragma unroll
  for (int step = 0; step < 16; ++step) {
    const int kk = step * 4 + khalf;         // even -> aligned b64 loads
    v2f A = *(const v2f*)(rowPtr + kk);      // Wh[row][kk], Wh[row][kk+1]
    v2f p = *(const v2f*)(bp + kk);
    v2f B;
    B[0] = p[0] * bmask;
    B[1] = p[1] * bmask;
    // 8 args: (neg_a, A, neg_b, B, c_mod, C, reuse_a, reuse_b)
    c = __builtin_amdgcn_wmma_f32_16x16x4_f32(false, A, false, B,
                                              (short)0, c, false, false);
  }

  // D[m][0] = s1[row16+m], D[m][1] = s2[row16+m].
  // N==0 lives in lanes {0,16}; N==1 in lanes {1,17}.
  const int half = lane >> 4;                // 0 -> rows v..v+7, 1 -> rows v+8..
  if (m < 2) {
    float* dst = (m == 0) ? s1 : s2;
    #pragma unroll
    for (int v = 0; v < 8; ++v)
      dst[row16 + v + 8 * half] = c[v];
  }
}

// ---------------------------------------------------------------------------
// Phase B: e[i][j] = leaky(s1[i] + s2[j]).  Pure store-bandwidth kernel
// (256 MB of output, ~11 us floor at 23.3 TB/s):
// one block per row, s1[row] uniform (scalar load), s2 cached float4 reads
// (32 KB, stays hot in L0/L2 across all rows), 8x float4 non-temporal stores
// per thread -> 512 B contiguous per wave, and NT keeps the 256 MB stream
// from thrashing the 192 MB L2.
// ---------------------------------------------------------------------------
__global__ __launch_bounds__(256) void gat_e_kernel(const float* __restrict__ s1,
                                                    const v4f* __restrict__ s2v,
                                                    v4f* __restrict__ out) {
  const int row = blockIdx.x;
  const float s = s1[row];                   // uniform -> s_load
  const int tid = threadIdx.x;
  v4f* outRow = out + (size_t)row * (N_ROWS / 4);

  #pragma unroll
  for (int it = 0; it < 8; ++it) {
    const int j4 = it * 256 + tid;           // 2048 float4 per row
    v4f t = s2v[j4];
    v4f e;
    #pragma unroll
    for (int k = 0; k < 4; ++k) {
      float v = s + t[k];
      e[k] = (v >= 0.0f) ? v : NEG_SLOPE * v;
    }
    __builtin_nontemporal_store(e, outRow + j4);  // global_store_b128, TH=NT
  }
}

extern "C" void kernel_launch(void* const* d_in, const int* in_sizes, int n_in,
                              void* d_out, int out_size, void* d_ws, size_t ws_size,
                              hipStream_t stream) {
  const float* Wh = (const float*)d_in[0];   // 8192 x 64 fp32
  const float* a  = (const float*)d_in[1];   // 128 x 1  fp32

  float* s1 = (float*)d_ws;                  // 8192 fp32
  float* s2 = s1 + N_ROWS;                   // 8192 fp32 (64 KB total)

  // Phase A: 512 waves (64 blocks x 8 waves), WMMA matvec.
  gat_s_kernel<<<64, 256, 0, stream>>>(Wh, a, s1, s2);

  // Phase B: 8192 blocks x 256 threads, bandwidth-bound outer sum + LeakyReLU.
  gat_e_kernel<<<N_ROWS, 256, 0, stream>>>(s1, (const v4f*)s2, (v4f*)d_out);
}